// SingleModalFactorization_85787676770631
// MI455X (gfx1250) — compile-verified
//
#include <hip/hip_runtime.h>

typedef __attribute__((ext_vector_type(2))) float v2f;
typedef __attribute__((ext_vector_type(4))) float v4f;
typedef __attribute__((ext_vector_type(8))) float v8f;

#define S_TOTAL   8192
#define C_CH      1024
#define K_COMP    4
#define N_OFF     9                 // d = -4..4
#define N_GROUPS  9                 // inner dim 36 = 9 groups of K=4
#define BLOCK     256
#define PAD       4                 // guard zeros each side of a component row
#define ROWSTRIDE (C_CH + 2 * PAD)  // 1032 floats per padded row

__global__ __launch_bounds__(BLOCK) void smf_shift_wmma_kernel(
    const float* __restrict__ comp,      // [4, 1024]
    const float* __restrict__ contrib,   // [8192, 4]
    const float* __restrict__ shift,     // [8192, 4]
    float* __restrict__ out)             // [8192, 1024]
{
    // Guard-padded component table: index k*1032 + 4 + c, c in [-4, 1027] always valid.
    __shared__ float lds_comp[K_COMP * ROWSTRIDE];   // 16.5 KB

    const int tid = threadIdx.x;

    // Zero the 32 guard cells (4 rows x 8) — one full wave, no sub-wave divergence.
    if (tid < 32) {
        const int k = tid >> 3, j = tid & 7;
        lds_comp[k * ROWSTRIDE + (j < PAD ? j : C_CH + j)] = 0.0f;
    }
    // Fill interiors with float4 copies (disjoint from guard cells — no race).
    #pragma unroll
    for (int i = 0; i < (K_COMP * C_CH / 4) / BLOCK; ++i) {
        const int idx = tid + i * BLOCK;   // 0..1023 float4 chunks
        const int k   = idx >> 8;          // 256 chunks per component row
        const int c4  = idx & 255;
        *(v4f*)(lds_comp + k * ROWSTRIDE + PAD + c4 * 4) =
            *(const v4f*)(comp + k * C_CH + c4 * 4);
    }
    __syncthreads();

    const int wave = tid >> 5;        // 0..7
    const int lane = tid & 31;
    const int half = lane >> 4;       // selects K pair within a group (A/B layout)
    const int n    = lane & 15;       // A: sample row M; B: channel col N
    const int s0   = (int)blockIdx.x * 16;   // sample-tile base (512 blocks)

    // Per-lane sample row data for the A (weight) fragments: sample s0+n.
    v4f sh4 = *(const v4f*)(shift   + (size_t)(s0 + n) * K_COMP);
    v4f ct4 = *(const v4f*)(contrib + (size_t)(s0 + n) * K_COMP);
    const float shv[4] = {sh4.x, sh4.y, sh4.z, sh4.w};
    const float ctv[4] = {ct4.x, ct4.y, ct4.z, ct4.w};

    // A fragments (once per sample tile): a[g][v] = W[m=n, r=4g+2*half+v],
    // W[s,r] = contrib[s,k] * max(1 - |d - shift[s,k]|, 0),  k=r/9, d=r%9-4.
    // LDS gather offsets (once per lane): off[g][v] = k*ROWSTRIDE + (d+4).
    v2f a[N_GROUPS];
    int off[N_GROUPS][2];
    #pragma unroll
    for (int g = 0; g < N_GROUPS; ++g) {
        #pragma unroll
        for (int v = 0; v < 2; ++v) {
            const int r   = 4 * g + 2 * half + v;
            const int k   = r / N_OFF;
            const int dp4 = r % N_OFF;            // d + 4, in 0..8
            float w = 1.0f - fabsf((float)(dp4 - 4) - shv[k]);
            a[g][v]   = fmaxf(w, 0.0f) * ctv[k];
            off[g][v] = k * ROWSTRIDE + dp4;
        }
    }

    // Each wave owns channel tiles ct = wave, wave+8, ... (64 tiles of 16 channels).
    for (int ct = wave; ct < C_CH / 16; ct += 8) {
        const int base = ct * 16 + n;             // c0 + n

        // Branch-free B gather: 18 unconditional ds_load_b32 (guards supply OOB zeros).
        v2f b[N_GROUPS];
        #pragma unroll
        for (int g = 0; g < N_GROUPS; ++g) {
            b[g][0] = lds_comp[off[g][0] + base];
            b[g][1] = lds_comp[off[g][1] + base];
        }

        // 9-deep f32 WMMA accumulation chain on the matrix pipe.
        v8f acc = {};
        #pragma unroll
        for (int g = 0; g < N_GROUPS; ++g) {
            acc = __builtin_amdgcn_wmma_f32_16x16x4_f32(
                /*neg_a=*/false, a[g], /*neg_b=*/false, b[g],
                /*c_mod=*/(short)0, acc, /*reuse_a=*/false, /*reuse_b=*/false);
        }

        // D layout: VGPR v, this lane -> out[s0 + v + 8*half, c0 + n].
        #pragma unroll
        for (int v = 0; v < 8; ++v) {
            out[(size_t)(s0 + v + 8 * half) * C_CH + base] = acc[v];
        }
    }
}

extern "C" void kernel_launch(void* const* d_in, const int* in_sizes, int n_in,
                              void* d_out, int out_size, void* d_ws, size_t ws_size,
                              hipStream_t stream) {
    // setup_inputs order: inputs (ignored by reference), components, contributions, shift
    const float* comp    = (const float*)d_in[1];   // [4, 1024]
    const float* contrib = (const float*)d_in[2];   // [8192, 4]
    const float* shift   = (const float*)d_in[3];   // [8192, 4]
    float*       out     = (float*)d_out;           // [8192, 1024]

    dim3 grid(S_TOTAL / 16);   // 512 sample tiles
    dim3 block(BLOCK);         // 8 waves; each wave computes 16x16 output tiles
    smf_shift_wmma_kernel<<<grid, block, 0, stream>>>(comp, contrib, shift, out);
    (void)in_sizes; (void)n_in; (void)out_size; (void)d_ws; (void)ws_size;
}